// VectorQuantizer_88012469829944
// MI455X (gfx1250) — compile-verified
//
#include <hip/hip_runtime.h>

// ---- problem constants (match reference) ----
#define N_PTS 131072
#define DIM   64
#define KCB   1024

#define ROWS_PER_WAVE  16
#define WAVES_PER_BLK  8
#define ROWS_PER_BLK   (ROWS_PER_WAVE * WAVES_PER_BLK)   // 128
#define NTILES         (KCB / 16)                        // 64 codebook tiles
#define KCHUNKS        (DIM / 4)                         // 16 WMMA K-steps

typedef __attribute__((ext_vector_type(2))) float v2f;
typedef __attribute__((ext_vector_type(4))) float v4f;
typedef __attribute__((ext_vector_type(8))) float v8f;

// zero the loss accumulator in workspace (keeps kernel_launch deterministic
// across graph replays -- harness does not re-poison between replays)
__global__ void vq_zero_ws(float* ws) {
    if (threadIdx.x == 0 && blockIdx.x == 0) ws[0] = 0.0f;
}

// finalize: loss = sum / (N*DIM)
__global__ void vq_finalize(const float* ws, float* out_loss) {
    if (threadIdx.x == 0 && blockIdx.x == 0)
        out_loss[0] = ws[0] * (1.0f / (float)(N_PTS * DIM));
}

// Main kernel: one wave32 owns 16 rows of x; whole codebook resident in LDS.
// Distances via V_WMMA_F32_16X16X4_F32 chained over DIM=64 (16 steps).
__global__ void __launch_bounds__(256, 1)
vq_main(const float* __restrict__ x,
        const float* __restrict__ emb,
        float* __restrict__ out_q,
        float* __restrict__ out_idx,
        float* __restrict__ loss_ws) {
    extern __shared__ float smem[];
    float* sEmb  = smem;               // KCB*DIM floats = 256 KB
    float* sNorm = smem + KCB * DIM;   // KCB floats     =   4 KB

    const int tid  = threadIdx.x;
    const int lane = tid & 31;
    const int wave = tid >> 5;

    // ---- stage 1: cooperative codebook load into LDS (float4 granularity) ----
    const v4f* embv = (const v4f*)emb;
    v4f* sEmbv = (v4f*)sEmb;
    #pragma unroll 4
    for (int i = tid; i < (KCB * DIM) / 4; i += 256) {
        sEmbv[i] = embv[i];
    }
    __syncthreads();

    // ---- stage 2: per-entry squared norms ----
    for (int e = tid; e < KCB; e += 256) {
        const float* row = sEmb + e * DIM;
        float s = 0.0f;
        #pragma unroll
        for (int d = 0; d < DIM; d += 4) {
            v4f v = *(const v4f*)(row + d);
            s += v.x * v.x + v.y * v.y + v.z * v.z + v.w * v.w;
        }
        sNorm[e] = s;
    }
    __syncthreads();

    // ---- stage 3: load A (x tile, 16x64) in WMMA f32 16x16x4 A-layout ----
    // A VGPR0: lanes 0-15 hold K=4k+0, lanes 16-31 hold K=4k+2
    // A VGPR1: lanes 0-15 hold K=4k+1, lanes 16-31 hold K=4k+3
    const int rowbase = blockIdx.x * ROWS_PER_BLK + wave * ROWS_PER_WAVE;
    const int m    = lane & 15;   // row within tile (A) / col entry (B)
    const int half = lane >> 4;   // K-pair selector

    const float* xrow = x + (size_t)(rowbase + m) * DIM + half * 2;
    v2f a[KCHUNKS];
    #pragma unroll
    for (int k = 0; k < KCHUNKS; ++k)
        a[k] = *(const v2f*)(xrow + 4 * k);   // global_load_b64

    // ---- stage 4: sweep all 64 codebook tiles with chained WMMA ----
    float best[8];
    int   bidx[8];
    #pragma unroll
    for (int j = 0; j < 8; ++j) { best[j] = 3.4e38f; bidx[j] = 0; }

    for (int t = 0; t < NTILES; ++t) {
        const int entry = t * 16 + m;
        // B (4x16) layout mirrors A: lane = column (codebook entry),
        // halves hold K pairs {0,1}/{2,3} of each 4-wide chunk
        const float* erow = sEmb + entry * DIM + half * 2;
        v2f b[KCHUNKS];
        #pragma unroll
        for (int k = 0; k < KCHUNKS; ++k)
            b[k] = *(const v2f*)(erow + 4 * k);   // ds_load_b64

        v8f c = {};
        #pragma unroll
        for (int k = 0; k < KCHUNKS; ++k)
            c = __builtin_amdgcn_wmma_f32_16x16x4_f32(
                    /*neg_a=*/false, a[k],
                    /*neg_b=*/false, b[k],
                    /*c_mod=*/(short)0, c,
                    /*reuse_a=*/false, /*reuse_b=*/false);

        // score = ||e||^2 - 2*dot  (||x||^2 is row-constant: irrelevant to argmin)
        const float en = sNorm[entry];
        #pragma unroll
        for (int j = 0; j < 8; ++j) {
            const float score = en - 2.0f * c[j];
            if (score < best[j]) { best[j] = score; bidx[j] = entry; }
        }
    }

    // ---- stage 5: cross-lane argmin (width-16: halves are rows j and j+8) ----
    // row m<8 ends up owned by lane m; row m>=8 by lane m+8
    int myRowIdx = 0;
    #pragma unroll
    for (int j = 0; j < 8; ++j) {
        float v = best[j];
        int   bi = bidx[j];
        #pragma unroll
        for (int off = 8; off >= 1; off >>= 1) {
            const float ov = __shfl_xor(v, off, 16);
            const int   oi = __shfl_xor(bi, off, 16);
            if (ov < v || (ov == v && oi < bi)) { v = ov; bi = oi; }
        }
        const int lo = __shfl(bi, 0, 32);   // row j
        const int hi = __shfl(bi, 16, 32);  // row j+8
        if (lane == j)      myRowIdx = lo;
        if (lane == 16 + j) myRowIdx = hi;
    }

    // ---- stage 6: gather quantized rows from LDS, fuse loss + outputs ----
    float lsum = 0.0f;
    const int col = lane * 2;   // 64 cols / 32 lanes = 2 cols per lane
    #pragma unroll 4
    for (int mm = 0; mm < 16; ++mm) {
        const int owner = (mm < 8) ? mm : (mm + 8);
        const int idx = __shfl(myRowIdx, owner, 32);
        if (lane == 0) out_idx[rowbase + mm] = (float)idx;

        const v2f e  = *(const v2f*)(sEmb + idx * DIM + col);
        const v2f xv = *(const v2f*)(x + (size_t)(rowbase + mm) * DIM + col);
        const float d0 = e.x - xv.x;
        const float d1 = e.y - xv.y;
        lsum += d0 * d0 + d1 * d1;
        // straight-through output == quantized values
        *(v2f*)(out_q + (size_t)(rowbase + mm) * DIM + col) = e;
    }

    // wave-level reduction of the loss partial, one atomic per wave
    #pragma unroll
    for (int off = 16; off >= 1; off >>= 1)
        lsum += __shfl_xor(lsum, off, 32);
    if (lane == 0) atomicAdd(loss_ws, lsum);
}

extern "C" void kernel_launch(void* const* d_in, const int* in_sizes, int n_in,
                              void* d_out, int out_size, void* d_ws, size_t ws_size,
                              hipStream_t stream) {
    const float* x   = (const float*)d_in[0];   // (N, 64) fp32
    const float* emb = (const float*)d_in[1];   // (K, 64) fp32

    float* out      = (float*)d_out;
    float* out_q    = out;                            // N*DIM floats
    float* out_idx  = out + (size_t)N_PTS * DIM;      // N floats (indices as f32)
    float* out_loss = out_idx + N_PTS;                // 1 float
    float* ws       = (float*)d_ws;

    vq_zero_ws<<<1, 64, 0, stream>>>(ws);

    const size_t smem_bytes = (size_t)(KCB * DIM + KCB) * sizeof(float); // 260 KB (<320 KB WGP LDS)
    vq_main<<<N_PTS / ROWS_PER_BLK, 256, smem_bytes, stream>>>(x, emb, out_q, out_idx, ws);

    vq_finalize<<<1, 64, 0, stream>>>(ws, out_loss);
}